// AutoEncoder_6425271075356
// MI455X (gfx1250) — compile-verified
//
#include <hip/hip_runtime.h>
#include <math.h>

// CDNA5 / gfx1250, wave32. Batch-1 MLP autoencoder + segmented softmax.
// Bandwidth-bound (~277 MB fp32 weights @ 23.3 TB/s => ~12us floor).
// GEMV done with V_WMMA_F32_16X16X4_F32: each WMMA consumes a 4x16 fp32
// weight tile (streamed with NT hint), x chunk lives in row 0 of A.

typedef float v2f __attribute__((ext_vector_type(2)));
typedef float v8f __attribute__((ext_vector_type(8)));

#define DS_   16384
#define H1_   2048
#define H2_   512
#define H3_   128
#define NSEG_ 1024

// ---------------- layer helpers ----------------

__global__ void init_vec_kernel(float* __restrict__ out,
                                const float* __restrict__ bias, int n) {
    int i = blockIdx.x * blockDim.x + threadIdx.x;
    if (i < n) out[i] = bias[i];
}

__global__ void tanh_inplace_kernel(float* __restrict__ v, int n) {
    int i = blockIdx.x * blockDim.x + threadIdx.x;
    if (i < n) v[i] = tanhf(v[i]);
}

// out[n0+c] += sum_{k in chunk} x[k] * W[k*N + n0+c]
// Grid: x = column-tile groups (8 waves/block, 1 tile each), y = K-chunk.
// A-matrix (16x4 f32, ISA 7.12.2): lane0 VGPR{0,1} = {K0,K1}, lane16 = {K2,K3},
//   rows M=1..15 zeroed -> D row 0 is the GEMV partial result.
// B-matrix (4x16 f32): VGPR0 lanes0-15 = row K0 (N=lane), lanes16-31 = row K2;
//   VGPR1 = rows K1 / K3 (mirrors the C/D half-split layout).
__global__ void gemv_wmma_kernel(const float* __restrict__ W,
                                 const float* __restrict__ x,
                                 float* __restrict__ out,
                                 int N, int chunkK) {
    const int wave = threadIdx.x >> 5;
    const int lane = threadIdx.x & 31;
    const int tile = blockIdx.x * 8 + wave;     // exact-fit grid, no guard
    const int n0   = tile << 4;
    const int c    = lane & 15;                 // column within tile
    const int half = lane >> 4;                 // 0: rows K0/K1, 1: rows K2/K3
    const int k0   = blockIdx.y * chunkK;
    const int k1   = k0 + chunkK;               // chunkK | K, multiple of 16
    const float* __restrict__ Wp = W + n0 + c;

    v8f acc = {};
    const float amask = (c == 0) ? 1.0f : 0.0f;

    for (int k = k0; k < k1; k += 16) {
#pragma unroll
        for (int u = 0; u < 4; ++u) {
            const int kk = k + 4 * u + 2 * half;
            v2f a, b;
            a.x = x[kk]     * amask;
            a.y = x[kk + 1] * amask;
            b.x = __builtin_nontemporal_load(Wp + (size_t)kk * N);
            b.y = __builtin_nontemporal_load(Wp + (size_t)(kk + 1) * N);
            acc = __builtin_amdgcn_wmma_f32_16x16x4_f32(
                false, a, false, b, (short)0, acc, false, false);
        }
    }

    // D row M=0 lives in VGPR0 of lanes 0..15 (N = lane).
    if (half == 0) atomicAdd(out + n0 + c, acc[0]);
}

// ---------------- segmented softmax ----------------

__device__ __forceinline__ unsigned enc_f32(float f) {
    unsigned b = __float_as_uint(f);
    return (b & 0x80000000u) ? ~b : (b ^ 0x80000000u);  // order-preserving
}
__device__ __forceinline__ float dec_f32(unsigned u) {
    unsigned b = (u & 0x80000000u) ? (u ^ 0x80000000u) : ~u;
    return __uint_as_float(b);
}

__global__ void seg_init_kernel(unsigned* __restrict__ segmax,
                                float* __restrict__ segsum) {
    int i = blockIdx.x * blockDim.x + threadIdx.x;
    if (i < NSEG_) {
        segmax[i] = enc_f32(-INFINITY);
        segsum[i] = 0.0f;
    }
}

__global__ void seg_max_kernel(const float* __restrict__ y,
                               const int* __restrict__ seg,
                               unsigned* __restrict__ segmax, int n) {
    int i = blockIdx.x * blockDim.x + threadIdx.x;
    if (i < n) atomicMax(segmax + seg[i], enc_f32(y[i]));
}

__global__ void seg_exp_kernel(const float* __restrict__ y,
                               const int* __restrict__ seg,
                               const unsigned* __restrict__ segmax,
                               float* __restrict__ segsum,
                               float* __restrict__ out, int n) {
    int i = blockIdx.x * blockDim.x + threadIdx.x;
    if (i < n) {
        int s = seg[i];
        float e = expf(y[i] - dec_f32(segmax[s]));
        out[i] = e;
        atomicAdd(segsum + s, e);
    }
}

__global__ void seg_norm_kernel(float* __restrict__ out,
                                const int* __restrict__ seg,
                                const float* __restrict__ segsum, int n) {
    int i = blockIdx.x * blockDim.x + threadIdx.x;
    if (i < n) out[i] = out[i] / segsum[seg[i]];
}

// ---------------- host side ----------------

static inline void launch_layer(const float* W, const float* x, float* out,
                                const float* bias, int K, int N, bool act,
                                hipStream_t s) {
    init_vec_kernel<<<(N + 255) / 256, 256, 0, s>>>(out, bias, N);
    const int chunkK = (K >= 512) ? 512 : K;   // split-K for MLP parallelism
    dim3 grid(N / 16 / 8, K / chunkK);         // 8 column tiles per block
    gemv_wmma_kernel<<<grid, 256, 0, s>>>(W, x, out, N, chunkK);
    if (act) tanh_inplace_kernel<<<(N + 255) / 256, 256, 0, s>>>(out, N);
}

extern "C" void kernel_launch(void* const* d_in, const int* in_sizes, int n_in,
                              void* d_out, int out_size, void* d_ws, size_t ws_size,
                              hipStream_t stream) {
    const float* x  = (const float*)d_in[0];
    const float* W1 = (const float*)d_in[1];  const float* b1 = (const float*)d_in[2];
    const float* W2 = (const float*)d_in[3];  const float* b2 = (const float*)d_in[4];
    const float* W3 = (const float*)d_in[5];  const float* b3 = (const float*)d_in[6];
    const float* W4 = (const float*)d_in[7];  const float* b4 = (const float*)d_in[8];
    const float* W5 = (const float*)d_in[9];  const float* b5 = (const float*)d_in[10];
    const float* W6 = (const float*)d_in[11]; const float* b6 = (const float*)d_in[12];
    const int* seg  = (const int*)d_in[13];
    float* out = (float*)d_out;

    float* ws = (float*)d_ws;
    float* h1 = ws;                 // 2048
    float* h2 = h1 + H1_;           // 512
    float* z  = h2 + H2_;           // 128
    float* h4 = z  + H3_;           // 512
    float* h5 = h4 + H2_;           // 2048
    float* y  = h5 + H1_;           // 16384
    unsigned* segmax = (unsigned*)(y + DS_);     // 1024
    float*    segsum = (float*)(segmax + NSEG_); // 1024

    // encoder
    launch_layer(W1, x,  h1, b1, DS_, H1_, true,  stream);
    launch_layer(W2, h1, h2, b2, H1_, H2_, true,  stream);
    launch_layer(W3, h2, z,  b3, H2_, H3_, false, stream);
    // decoder
    launch_layer(W4, z,  h4, b4, H3_, H2_, true,  stream);
    launch_layer(W5, h4, h5, b5, H2_, H1_, true,  stream);
    launch_layer(W6, h5, y,  b6, H1_, DS_, false, stream);

    // segmented softmax (contiguous segments, 1024 of them)
    seg_init_kernel<<<(NSEG_ + 255) / 256, 256, 0, stream>>>(segmax, segsum);
    seg_max_kernel <<<(DS_ + 255) / 256, 256, 0, stream>>>(y, seg, segmax, DS_);
    seg_exp_kernel <<<(DS_ + 255) / 256, 256, 0, stream>>>(y, seg, segmax, segsum, out, DS_);
    seg_norm_kernel<<<(DS_ + 255) / 256, 256, 0, stream>>>(out, seg, segsum, DS_);
}